// AUTOGCNLayer_65919158059662
// MI455X (gfx1250) — compile-verified
//
#include <hip/hip_runtime.h>
#include <hip/hip_bf16.h>

typedef __attribute__((ext_vector_type(2))) float v2f;
typedef __attribute__((ext_vector_type(8))) float v8f;
typedef __attribute__((ext_vector_type(4))) unsigned int v4u;
typedef __attribute__((ext_vector_type(8))) int v8i;
typedef __attribute__((ext_vector_type(4))) int v4i;

#define NN 100000
#define EE 3200000
#define DD 256
#define KK 8
#define EPSF 1e-09f

#if __has_builtin(__builtin_amdgcn_tensor_load_to_lds)
#define HAVE_TDM 1
#else
#define HAVE_TDM 0
#endif

// ---------------------------------------------------------------- degree
__global__ void deg_kernel(const int* __restrict__ dst, float* __restrict__ deg) {
    int e = blockIdx.x * 256 + threadIdx.x;
    if (e < EE) unsafeAtomicAdd(&deg[dst[e]], 1.0f);
}

__global__ void norm_kernel(const float* __restrict__ deg, float* __restrict__ nrm) {
    int i = blockIdx.x * 256 + threadIdx.x;
    if (i < NN) nrm[i] = __frsqrt_rn(fmaxf(deg[i], 1.0f));
}

// buf1 = feature * norm[row]   (float4 over N*D)
__global__ void scale_kernel(const float* __restrict__ x, const float* __restrict__ nrm,
                             float* __restrict__ out) {
    long long i = (long long)blockIdx.x * 256 + threadIdx.x;      // float4 units
    int row = (int)(i >> 6);                                       // D/4 = 64 float4 per row
    float s = nrm[row];
    const float4* x4 = (const float4*)x;
    float4* o4 = (float4*)out;
    float4 v = x4[i];
    v.x *= s; v.y *= s; v.z *= s; v.w *= s;
    o4[i] = v;
}

// scatter-add SpMM: out[dst] += in[src]; 4 edges per 256-thread block, float4 per 64-lane chunk
__global__ void scatter_kernel(const int* __restrict__ src, const int* __restrict__ dst,
                               const float* __restrict__ in, float* __restrict__ out) {
    int e = blockIdx.x * 4 + (threadIdx.x >> 6);
    int q = threadIdx.x & 63;
    if (e < EE) {
        long long s = src[e], d = dst[e];
        const float4* in4 = (const float4*)in;
        float4 v = in4[s * 64 + q];
        float* o = out + d * DD + q * 4;
        unsafeAtomicAdd(o + 0, v.x);
        unsafeAtomicAdd(o + 1, v.y);
        unsafeAtomicAdd(o + 2, v.z);
        unsafeAtomicAdd(o + 3, v.w);
    }
}

// h = agg*norm (in place); hs = h*norm (input for second hop)
__global__ void hscale_kernel(float* __restrict__ agg, const float* __restrict__ nrm,
                              float* __restrict__ hs) {
    long long i = (long long)blockIdx.x * 256 + threadIdx.x;
    int row = (int)(i >> 6);
    float s = nrm[row];
    float4* a4 = (float4*)agg;
    float4* h4 = (float4*)hs;
    float4 v = a4[i];
    v.x *= s; v.y *= s; v.z *= s; v.w *= s;
    a4[i] = v;
    v.x *= s; v.y *= s; v.z *= s; v.w *= s;
    h4[i] = v;
}

__global__ void h1scale_kernel(float* __restrict__ agg, const float* __restrict__ nrm) {
    long long i = (long long)blockIdx.x * 256 + threadIdx.x;
    int row = (int)(i >> 6);
    float s = nrm[row];
    float4* a4 = (float4*)agg;
    float4 v = a4[i];
    v.x *= s; v.y *= s; v.z *= s; v.w *= s;
    a4[i] = v;
}

// fold gamma/alpha into 6 scalars
__global__ void coef_kernel(const float* __restrict__ gl, const float* __restrict__ gm,
                            const float* __restrict__ gh, float* __restrict__ c) {
    if (threadIdx.x == 0 && blockIdx.x == 0) {
        float c0 = 0, c1 = 0, c2 = 0, c3 = 0, c4 = 0, c5 = 0;
        for (int k = 0; k < KK; ++k) {
            float a = -EPSF + (float)k * (1.0f + 2.0f * EPSF) / (float)(KK - 1);
            float l = fmaxf(gl[k], 0.0f);
            float m = fmaxf(gm[k], 0.0f);
            float h = fmaxf(gh[k], 0.0f);
            c0 += a * l;          c1 += (1.0f - a) * l;
            c2 += -a * h;         c3 += (1.0f - a) * h;
            c4 += m;              c5 += a * m;
        }
        c[0] = c0; c[1] = c1; c[2] = c2; c[3] = c3; c[4] = c4; c[5] = c5;
    }
}

// Wt[k*D + n] = W[n*D + k]
__global__ void transpose_kernel(const float* __restrict__ W0, const float* __restrict__ W1,
                                 const float* __restrict__ W2, float* __restrict__ T0,
                                 float* __restrict__ T1, float* __restrict__ T2) {
    int b = blockIdx.x;
    int mat = b >> 8;
    int k = b & 255;
    int n = threadIdx.x;
    const float* W = (mat == 0) ? W0 : (mat == 1) ? W1 : W2;
    float* T = (mat == 0) ? T0 : (mat == 1) ? T1 : T2;
    T[(long long)k * DD + n] = W[(long long)n * DD + k];
}

__device__ __forceinline__ float sigmoidf(float x) {
    // fast: v_exp + v_add + v_rcp (no IEEE divide sequence)
    return __builtin_amdgcn_rcpf(1.0f + __expf(-x));
}

#define LDS_STRIDE 260   // 256 + 4 pad (TDM pad_amount=4 DWORDs) kills bank conflicts
#define SLAB_FLOATS (16 * LDS_STRIDE)
#define SLAB_BYTES (SLAB_FLOATS * 4)

#if HAVE_TDM
// TDM: async 2D tile load (16 rows x 256 f32) from a [100000 x 256] tensor into LDS,
// with hardware row padding of 4 DWORDs every 256 DWORDs (-> 260-float LDS row stride).
__device__ __forceinline__ void tdm_load_tile(const float* gsrc, unsigned int lds_off) {
    unsigned long long ga = (unsigned long long)(uintptr_t)gsrc;
    v4u g0;
    g0.x = 0x1u;                                       // count=1, user mode
    g0.y = lds_off;                                    // lds_addr
    g0.z = (unsigned int)(ga & 0xFFFFFFFFu);           // global_addr[31:0]
    g0.w = (unsigned int)((ga >> 32) & 0x01FFFFFFu)    // global_addr[56:32]
         | (2u << 30);                                 // type=2 ("image")
    v8i g1;
    g1[0] = (int)((2u << 16)        // data_size = 4 bytes
                | (1u << 20)        // pad_enable
                | (7u << 22)        // pad_interval = 256 DWORDs
                | (3u << 25));      // pad_amount  = 4 DWORDs
    g1[1] = (int)(256u << 16);                         // tensor_dim0[15:0]=256
    g1[2] = (int)(((unsigned int)(NN & 0xFFFF)) << 16);// tensor_dim1[15:0]
    g1[3] = (int)((256u << 16) | ((NN >> 16) & 0xFFFF));// tile_dim0=256 | tensor_dim1[31:16]
    g1[4] = 16;                                        // tile_dim1 = 16 rows
    g1[5] = 256;                                       // tensor_dim0_stride
    g1[6] = 0;
    g1[7] = 0;
    v4i z4 = {0, 0, 0, 0};
#if defined(__clang_major__) && __clang_major__ >= 23
    v8i z8 = {0, 0, 0, 0, 0, 0, 0, 0};
    __builtin_amdgcn_tensor_load_to_lds(g0, g1, z4, z4, z8, 0);
#else
    __builtin_amdgcn_tensor_load_to_lds(g0, g1, z4, z4, 0);
#endif
}
#endif

// Fused triple-GEMM (fp32 WMMA) + sequential sigmoid cross-gating + bias + snorm + relu.
// Block = 16 node rows x 256 cols; 8 waves, each wave owns two 16x16 output col-tiles.
__global__ void __launch_bounds__(256)
gemm_gate_kernel(const float* __restrict__ feat, const float* __restrict__ h,
                 const float* __restrict__ h1, const float* __restrict__ BtL,
                 const float* __restrict__ BtM, const float* __restrict__ BtH,
                 const float* __restrict__ c, const float* __restrict__ bias,
                 const float* __restrict__ snorm, float* __restrict__ out) {
    // 3 TDM-written slabs + 256-float dummy region (written by threads so the
    // compiler cannot prove this global store-free and fold its loads to undef).
    __shared__ float lds[3 * SLAB_FLOATS + 256];

    const int tid = threadIdx.x;
    const long long rowbase = (long long)blockIdx.x * 16;

    const float c0 = c[0], c1 = c[1], c2 = c[2], c3 = c[3], c4 = c[4], c5 = c[5];

#if HAVE_TDM
    // Async-tensor staging: one wave issues 3 TDM tile loads (TDM ignores EXEC, so
    // issuance must be branch-guarded to a single wave), waits TENSORcnt, then barrier.
    if (tid < 32) {
        tdm_load_tile(feat + rowbase * DD, 0u);
        tdm_load_tile(h    + rowbase * DD, (unsigned int)SLAB_BYTES);
        tdm_load_tile(h1   + rowbase * DD, (unsigned int)(2 * SLAB_BYTES));
        __builtin_amdgcn_s_wait_tensorcnt(0);
    }
    // Defeat undef-folding of LDS loads (TDM writes are invisible to alias
    // analysis): (1) a real store into the array, (2) escape the array's
    // address into a memory-clobbering asm that may read/write it.
    lds[3 * SLAB_FLOATS + tid] = 0.0f;
    asm volatile("" :: "v"((void*)lds) : "memory");
    __syncthreads();
#else
    #pragma unroll
    for (int it = 0; it < 16; ++it) {
        int idx = it * 256 + tid;
        int r = idx >> 8;
        int k = idx & 255;
        long long g = (rowbase + r) * DD + k;
        lds[0 * SLAB_FLOATS + r * LDS_STRIDE + k] = feat[g];
        lds[1 * SLAB_FLOATS + r * LDS_STRIDE + k] = h[g];
        lds[2 * SLAB_FLOATS + r * LDS_STRIDE + k] = h1[g];
    }
    __syncthreads();
#endif

    const float* Lf  = lds;                       // feat slab
    const float* Lh  = lds + SLAB_FLOATS;         // h slab
    const float* Lh1 = lds + 2 * SLAB_FLOATS;     // h1 slab

    const int lane = tid & 31;
    const int wave = tid >> 5;
    const int m = lane & 15;                 // A row / B col within tile
    const int khalf = (lane >> 4) << 1;      // 0 or 2
    const int ct0 = wave * 2, ct1 = wave * 2 + 1;
    const int n0 = ct0 * 16 + m, n1 = ct1 * 16 + m;
    const int arow = m * LDS_STRIDE;

    v8f aL0 = {}, aL1 = {}, aH0 = {}, aH1 = {}, aM0 = {}, aM1 = {};

    for (int kb = 0; kb < DD; kb += 4) {
        int kk = kb + khalf;
        v2f ff  = { Lf[arow + kk],  Lf[arow + kk + 1] };
        v2f hh  = { Lh[arow + kk],  Lh[arow + kk + 1] };
        v2f hh1 = { Lh1[arow + kk], Lh1[arow + kk + 1] };
        // on-the-fly alpha/gamma combination (co-executes with WMMA)
        v2f fL = c0 * hh + c1 * ff;          // low
        v2f fH = c2 * hh + c3 * ff;          // high
        v2f fM = c4 * hh1 - c5 * ff;         // mid

        long long r0 = (long long)kk * DD, r1 = (long long)(kk + 1) * DD;
        v2f bL0 = { BtL[r0 + n0], BtL[r1 + n0] };
        v2f bL1 = { BtL[r0 + n1], BtL[r1 + n1] };
        v2f bM0 = { BtM[r0 + n0], BtM[r1 + n0] };
        v2f bM1 = { BtM[r0 + n1], BtM[r1 + n1] };
        v2f bH0 = { BtH[r0 + n0], BtH[r1 + n0] };
        v2f bH1 = { BtH[r0 + n1], BtH[r1 + n1] };

        aL0 = __builtin_amdgcn_wmma_f32_16x16x4_f32(false, fL, false, bL0, (short)0, aL0, false, false);
        aL1 = __builtin_amdgcn_wmma_f32_16x16x4_f32(false, fL, false, bL1, (short)0, aL1, false, false);
        aH0 = __builtin_amdgcn_wmma_f32_16x16x4_f32(false, fH, false, bH0, (short)0, aH0, false, false);
        aH1 = __builtin_amdgcn_wmma_f32_16x16x4_f32(false, fH, false, bH1, (short)0, aH1, false, false);
        aM0 = __builtin_amdgcn_wmma_f32_16x16x4_f32(false, fM, false, bM0, (short)0, aM0, false, false);
        aM1 = __builtin_amdgcn_wmma_f32_16x16x4_f32(false, fM, false, bM1, (short)0, aM1, false, false);
    }

    // epilogue: C/D layout — VGPR r holds row (r + 8*(lane>>4)), col = lane&15 (+ 16*ct)
    const int rofs = (lane >> 4) << 3;
    float bias0 = bias[n0];
    float bias1 = bias[n1];

    #pragma unroll
    for (int r = 0; r < 8; ++r) {
        long long row = rowbase + r + rofs;
        float sn = snorm[row];
        {
            float l = aL0[r], hi = aH0[r], md = aM0[r];
            l = l * sigmoidf(hi + md);
            md = md * sigmoidf(l + hi);
            hi = hi * sigmoidf(md + l);
            float o = (l + md + hi + bias0) * sn;
            out[row * DD + n0] = fmaxf(o, 0.0f);
        }
        {
            float l = aL1[r], hi = aH1[r], md = aM1[r];
            l = l * sigmoidf(hi + md);
            md = md * sigmoidf(l + hi);
            hi = hi * sigmoidf(md + l);
            float o = (l + md + hi + bias1) * sn;
            out[row * DD + n1] = fmaxf(o, 0.0f);
        }
    }
}

extern "C" void kernel_launch(void* const* d_in, const int* in_sizes, int n_in,
                              void* d_out, int out_size, void* d_ws, size_t ws_size,
                              hipStream_t stream) {
    const float* feature = (const float*)d_in[0];
    const float* snorm   = (const float*)d_in[1];
    const float* W_low   = (const float*)d_in[2];
    const float* W_mid   = (const float*)d_in[3];
    const float* W_high  = (const float*)d_in[4];
    const float* g_low   = (const float*)d_in[5];
    const float* g_mid   = (const float*)d_in[6];
    const float* g_high  = (const float*)d_in[7];
    const float* bias    = (const float*)d_in[8];
    const int*   src     = (const int*)d_in[9];
    const int*   dst     = (const int*)d_in[10];
    float* out = (float*)d_out;

    const long long N = NN, D = DD;

    float* ws   = (float*)d_ws;
    float* deg  = ws;                    // N
    float* nrm  = deg + N;               // N
    float* cbuf = nrm + N;               // 16
    float* WtL  = cbuf + 16;             // D*D
    float* WtM  = WtL + D * D;
    float* WtH  = WtM + D * D;
    float* buf1 = WtH + D * D;           // N*D  (scaled features / scaled h)
    float* buf2 = buf1 + N * D;          // N*D  (agg1 -> h)
    float* buf3 = buf2 + N * D;          // N*D  (agg2 -> h1)

    hipMemsetAsync(deg,  0, N * sizeof(float), stream);
    hipMemsetAsync(buf2, 0, N * D * sizeof(float), stream);
    hipMemsetAsync(buf3, 0, N * D * sizeof(float), stream);

    coef_kernel<<<1, 32, 0, stream>>>(g_low, g_mid, g_high, cbuf);
    transpose_kernel<<<3 * DD, DD, 0, stream>>>(W_low, W_mid, W_high, WtL, WtM, WtH);

    // hop 1
    deg_kernel<<<(EE + 255) / 256, 256, 0, stream>>>(dst, deg);
    norm_kernel<<<(NN + 255) / 256, 256, 0, stream>>>(deg, nrm);
    scale_kernel<<<(int)((N * D / 4 + 255) / 256), 256, 0, stream>>>(feature, nrm, buf1);
    scatter_kernel<<<(EE + 3) / 4, 256, 0, stream>>>(src, dst, buf1, buf2);
    hscale_kernel<<<(int)((N * D / 4 + 255) / 256), 256, 0, stream>>>(buf2, nrm, buf1);

    // hop 2
    scatter_kernel<<<(EE + 3) / 4, 256, 0, stream>>>(src, dst, buf1, buf3);
    h1scale_kernel<<<(int)((N * D / 4 + 255) / 256), 256, 0, stream>>>(buf3, nrm);

    // fused triple GEMM + gating + bias + snorm + relu
    gemm_gate_kernel<<<NN / 16, 256, 0, stream>>>(feature, buf2, buf3, WtL, WtM, WtH,
                                                  cbuf, bias, snorm, out);
}